// Attention_32495722561949
// MI455X (gfx1250) — compile-verified
//
#include <hip/hip_runtime.h>
#include <cmath>

typedef __attribute__((ext_vector_type(2))) float v2f;
typedef __attribute__((ext_vector_type(8))) float v8f;

#define H 512
#define B 64
#define S 2048

// Branch-free tanh: hardware v_tanh_f32 if the builtin exists, else exp-based.
__device__ __forceinline__ float fast_tanh(float x) {
#if __has_builtin(__builtin_amdgcn_tanhf)
    return __builtin_amdgcn_tanhf(x);
#else
    // tanh(x) = sign(x) * (1 - e) / (1 + e),  e = exp(-2|x|)
    float ax = __builtin_fabsf(x);
    float e  = __expf(-2.0f * ax);            // v_exp_f32 path, no branches
    float r  = (1.0f - e) * __frcp_rn(1.0f + e);
    return __builtin_copysignf(r, x);
#endif
}

// ---------------- Kernel 1: q = query @ Wq^T + bq  (tiny) ----------------
__global__ __launch_bounds__(256)
void qproj_kernel(const float* __restrict__ query,
                  const float* __restrict__ Wq,
                  const float* __restrict__ bq,
                  float* __restrict__ q_out) {
    int tid = blockIdx.x * blockDim.x + threadIdx.x;   // 0 .. B*H-1
    int b = tid >> 9;            // / H
    int h = tid & (H - 1);       // % H
    const float4* qr = reinterpret_cast<const float4*>(query + (size_t)b * H);
    const float4* wr = reinterpret_cast<const float4*>(Wq + (size_t)h * H);
    float acc = 0.f;
#pragma unroll 4
    for (int k = 0; k < H / 4; ++k) {
        float4 a = qr[k];
        float4 w = wr[k];
        acc += a.x * w.x + a.y * w.y + a.z * w.z + a.w * w.w;
    }
    q_out[tid] = acc + bq[h];
}

// ------- Kernel 2: fused  ref_t = Wr@ref^T + br ;  logits = V.tanh(q+ref_t) -------
// Grid: B * (S/32) workgroups of 256 threads (8 waves).
// Wave w owns o in [w*64, w*64+64) as 4 o-tiles of 16, and both 16-wide s-subtiles.
__global__ __launch_bounds__(256)
void fused_kernel(const float* __restrict__ ref,    // [B,S,H]
                  const float* __restrict__ Wr,     // [H,H] row-major [o][h]
                  const float* __restrict__ br,     // [H]
                  const float* __restrict__ Vv,     // [H]
                  const float* __restrict__ qvec,   // [B,H]
                  float* __restrict__ ref_t,        // [B,H,S]
                  float* __restrict__ logits) {     // [B,S]
    __shared__ float lds_log[32];

    const int SW  = S / 32;
    const int b   = blockIdx.x / SW;
    const int s0  = (blockIdx.x % SW) * 32;
    const int wave = threadIdx.x >> 5;
    const int lane = threadIdx.x & 31;
    const int row  = lane & 15;      // A: M row / B: N column
    const int kh   = lane >> 4;      // which K-pair this half-wave holds

    if (threadIdx.x < 32) lds_log[threadIdx.x] = 0.f;
    __syncthreads();

    const int o_base = wave * 64;

    // Per-lane base pointers (K-pair offset kh*2 baked in; loads are float2 / b64).
    const float* Abase = Wr  + (size_t)(o_base + row) * H + kh * 2;
    const float* Bbase = ref + ((size_t)b * S + s0 + row) * H + kh * 2;

    v8f acc[4][2];
#pragma unroll
    for (int ot = 0; ot < 4; ++ot)
#pragma unroll
        for (int n = 0; n < 2; ++n)
            acc[ot][n] = {};

#pragma unroll 2
    for (int k = 0; k < H; k += 4) {
        v2f amat[4];
#pragma unroll
        for (int ot = 0; ot < 4; ++ot)
            amat[ot] = *reinterpret_cast<const v2f*>(Abase + (size_t)ot * 16 * H + k);
        v2f bmat[2];
#pragma unroll
        for (int n = 0; n < 2; ++n)
            bmat[n] = *reinterpret_cast<const v2f*>(Bbase + (size_t)n * 16 * H + k);

#pragma unroll
        for (int ot = 0; ot < 4; ++ot)
#pragma unroll
            for (int n = 0; n < 2; ++n)
                acc[ot][n] = __builtin_amdgcn_wmma_f32_16x16x4_f32(
                    false, amat[ot], false, bmat[n],
                    (short)0, acc[ot][n], false, false);
    }

    // Epilogue: +br, store ref_t, accumulate V*tanh(q + val) per s column.
    float part0 = 0.f, part1 = 0.f;
    const float* qb = qvec + (size_t)b * H;
#pragma unroll
    for (int ot = 0; ot < 4; ++ot) {
#pragma unroll
        for (int i = 0; i < 8; ++i) {
            // C/D layout: VGPR i holds M=i (lanes 0-15) or M=i+8 (lanes 16-31).
            const int o   = o_base + ot * 16 + i + kh * 8;
            const float bro = br[o];
            const float qo  = qb[o];
            const float vo  = Vv[o];

            float v0 = acc[ot][0][i] + bro;
            float v1 = acc[ot][1][i] + bro;
            const size_t rowbase = ((size_t)b * H + o) * S + s0 + row;
            ref_t[rowbase]      = v0;
            ref_t[rowbase + 16] = v1;
            part0 += vo * fast_tanh(qo + v0);
            part1 += vo * fast_tanh(qo + v1);
        }
    }
    atomicAdd(&lds_log[row],      part0);
    atomicAdd(&lds_log[16 + row], part1);
    __syncthreads();
    if (threadIdx.x < 32)
        logits[(size_t)b * S + s0 + threadIdx.x] = lds_log[threadIdx.x];
}

// ------------------------------ launcher ------------------------------
extern "C" void kernel_launch(void* const* d_in, const int* in_sizes, int n_in,
                              void* d_out, int out_size, void* d_ws, size_t ws_size,
                              hipStream_t stream) {
    (void)in_sizes; (void)n_in; (void)out_size; (void)ws_size;
    const float* query = (const float*)d_in[0];   // [B,H]
    const float* ref   = (const float*)d_in[1];   // [B,S,H]
    const float* Wq    = (const float*)d_in[2];   // [H,H]
    const float* bq    = (const float*)d_in[3];   // [H]
    const float* Wr    = (const float*)d_in[4];   // [H,H]
    const float* br    = (const float*)d_in[5];   // [H]
    const float* V     = (const float*)d_in[6];   // [H]

    float* out    = (float*)d_out;
    float* ref_t  = out;                           // B*H*S floats
    float* logits = out + (size_t)B * H * S;       // B*S floats
    float* qws    = (float*)d_ws;                  // B*H floats scratch

    qproj_kernel<<<(B * H) / 256, 256, 0, stream>>>(query, Wq, bq, qws);
    fused_kernel<<<B * (S / 32), 256, 0, stream>>>(ref, Wr, br, V, qws, ref_t, logits);
}